// Mol2MSLoss_48387101557303
// MI455X (gfx1250) — compile-verified
//
#include <hip/hip_runtime.h>

// Mol2MS soft-Jaccard loss for MI455X (gfx1250, wave32).
//
// min_j |a-b| via squared distance as a rank-3 contraction on
// V_WMMA_F32_16X16X4_F32:   (a-b)^2 = [a^2, -2a, 1, 0] . [1, b, b^2, 0]
// One WMMA = one 16x16 tile of squared distances; raw v_min_num_f32
// (inline asm, no IEEE canonicalization — inputs are finite) accumulates
// the running min over j-tiles. B operands are pre-packed per (j-tile,lane)
// in LDS and software-pipelined (register double-buffer) so the ds_load
// latency is hidden behind the previous tile's 4 WMMA + 32 min chain.
// All register-array loops are force-unrolled so acc[][]/A[] stay in
// static VGPRs (round-2 lesson: rolled loops -> v_movrels disaster).

typedef __attribute__((ext_vector_type(2))) float v2f;
typedef __attribute__((ext_vector_type(8))) float v8f;

#define B_DIM 64
#define S_DIM 2048
#define JT_COUNT (S_DIM / 16)  // 128
#define INV_THETA 10.0f        // 1 / SOFT_MATCH_THRESHOLD
#define EPSF 1e-8f
#define LEN_W 0.01f
#define BLOCKS_PER_BATCH 4
#define WAVES_PER_BLOCK 8
#define ITILES_PER_WAVE (JT_COUNT / (BLOCKS_PER_BATCH * WAVES_PER_BLOCK)) // = 4

__device__ __forceinline__ float fmin_raw(float a, float b) {
    float r;
    asm("v_min_num_f32 %0, %1, %2" : "=v"(r) : "v"(a), "v"(b));
    return r;
}

// ---------------- workspace init ----------------
__global__ void init_ws_kernel(float* inter, int* pcnt, int* tcnt) {
    int t = threadIdx.x;
    if (t < B_DIM) {
        inter[t] = 0.0f;
        pcnt[t] = 0;
        tcnt[t] = 0;
    }
}

// ---------------- length counts (ballot + popcount) ----------------
__global__ __launch_bounds__(256) void count_kernel(
    const float* __restrict__ pred_mz, const float* __restrict__ mz,
    int* __restrict__ pcnt, int* __restrict__ tcnt) {
    int idx = blockIdx.x * 256 + threadIdx.x;               // < B*S; S%32==0 so
    unsigned long long m1 = __ballot(pred_mz[idx] >= 0.0f); // a wave never
    unsigned long long m2 = __ballot(mz[idx] >= 0.0f);      // straddles a batch
    if ((threadIdx.x & 31) == 0) {
        int b = idx / S_DIM;
        atomicAdd(&pcnt[b], (int)__popcll(m1));
        atomicAdd(&tcnt[b], (int)__popcll(m2));
    }
}

// ---------------- main WMMA kernel ----------------
__global__ __launch_bounds__(256) void soft_match_kernel(
    const float* __restrict__ pred_mz, const float* __restrict__ mz,
    const float* __restrict__ inten, float* __restrict__ inter) {
    // Pre-packed B operand per (j-tile, lane): the exact float2 each lane
    // feeds to the WMMA B input. f32 4x16 B layout:
    //   lanes 0-15  (N=lane):    v0 = K0 = 1.0,   v1 = K1 = b
    //   lanes 16-31 (N=lane-16): v0 = K2 = b^2,   v1 = K3 = 0
    __shared__ v2f sBB[JT_COUNT * 32];   // 32 KB

    const int b    = blockIdx.x >> 2;    // batch
    const int sub  = blockIdx.x & 3;     // sub-block within batch
    const int tid  = threadIdx.x;
    const int lane = tid & 31;
    const int wave = tid >> 5;
    const int l16  = lane & 15;
    const int hi   = lane >> 4;          // 0: lanes 0-15, 1: lanes 16-31

    // Stage packed B operands once per block (mz row lives in L2, 8 KB).
    for (int e = tid; e < JT_COUNT * 32; e += 256) {
        int jt = e >> 5;
        int ln = e & 31;
        float t = mz[(size_t)b * S_DIM + jt * 16 + (ln & 15)];
        v2f p;
        p.x = (ln < 16) ? 1.0f : t * t;
        p.y = (ln < 16) ? t : 0.0f;
        sBB[e] = p;
    }
    __syncthreads();

    const int gw = sub * WAVES_PER_BLOCK + wave;   // 0..31 within batch

    // A operands: row i = [a^2, -2a, 1, 0].
    // f32 16x4 A layout: lanes 0-15 -> K0 (v0), K1 (v1); lanes 16-31 -> K2, K3.
    v2f A[ITILES_PER_WAVE];
    int itile[ITILES_PER_WAVE];
#pragma unroll
    for (int k = 0; k < ITILES_PER_WAVE; ++k) {
        itile[k] = gw * ITILES_PER_WAVE + k;
        float a = pred_mz[(size_t)b * S_DIM + itile[k] * 16 + l16];
        A[k].x = hi ? 1.0f : a * a;
        A[k].y = hi ? 0.0f : -2.0f * a;
    }

    v8f acc[ITILES_PER_WAVE];
#pragma unroll
    for (int k = 0; k < ITILES_PER_WAVE; ++k)
#pragma unroll
        for (int r = 0; r < 8; ++r) acc[k][r] = 3.0e38f;

    const v8f czero = {};

    // One j-tile of work: 4 WMMA + 32 raw v_min.
    auto tile_body = [&](v2f Bmv) {
#pragma unroll
        for (int k = 0; k < ITILES_PER_WAVE; ++k) {
            v8f d = __builtin_amdgcn_wmma_f32_16x16x4_f32(
                false, A[k], false, Bmv, (short)0, czero, false, false);
#pragma unroll
            for (int r = 0; r < 8; ++r) acc[k][r] = fmin_raw(acc[k][r], d[r]);
        }
    };

    // Software-pipelined hot loop: ds_load for jt+1 issued before consuming
    // jt, so s_wait_dscnt covers a load that is a full iteration old.
    v2f Bm = sBB[lane];  // jt = 0
#pragma unroll 1
    for (int jt = 0; jt < JT_COUNT - 1; ++jt) {
        v2f Bnext = sBB[((jt + 1) << 5) + lane];
        tile_body(Bm);
        Bm = Bnext;
    }
    tile_body(Bm);       // peeled last tile

    // Min over the 16 columns held across each 16-lane half (xor masks < 16
    // stay inside a half on wave32).
#pragma unroll
    for (int k = 0; k < ITILES_PER_WAVE; ++k) {
#pragma unroll
        for (int r = 0; r < 8; ++r) {
            float v = acc[k][r];
            v = fmin_raw(v, __shfl_xor(v, 1, 32));
            v = fmin_raw(v, __shfl_xor(v, 2, 32));
            v = fmin_raw(v, __shfl_xor(v, 4, 32));
            v = fmin_raw(v, __shfl_xor(v, 8, 32));
            acc[k][r] = v;
        }
    }

    // Lane 0 holds rows 0..7 of each tile (VGPR r -> row r);
    // lane 16 holds rows 8..15 (VGPR r -> row r+8).
    if (l16 == 0) {
#pragma unroll
        for (int k = 0; k < ITILES_PER_WAVE; ++k) {
            int base = itile[k] * 16 + hi * 8;
            float local = 0.0f;
#pragma unroll
            for (int r = 0; r < 8; ++r) {
                float d2 = fmaxf(acc[k][r], 0.0f);  // clamp rounding negatives
                float sm = __expf(-__builtin_sqrtf(d2) * INV_THETA);
                local += sm * inten[(size_t)b * S_DIM + base + r];
            }
            atomicAdd(&inter[b], local);
        }
    }
}

// ---------------- finalize: scalar loss ----------------
__global__ void finalize_kernel(const float* __restrict__ inter,
                                const int* __restrict__ pcnt,
                                const int* __restrict__ tcnt,
                                float* __restrict__ out) {
    __shared__ float red[B_DIM];
    int t = threadIdx.x;                    // 64 threads
    float soft_union = (float)(2 * S_DIM);  // S + S
    float jac = (inter[t] + EPSF) / (soft_union + EPSF);
    int dl = pcnt[t] - tcnt[t];
    float pen = fabsf((float)dl) * LEN_W;
    red[t] = (1.0f - jac) + pen;
    __syncthreads();
    for (int off = 32; off > 0; off >>= 1) {
        if (t < off) red[t] += red[t + off];
        __syncthreads();
    }
    if (t == 0) out[0] = red[0] * (1.0f / (float)B_DIM);
}

extern "C" void kernel_launch(void* const* d_in, const int* in_sizes, int n_in,
                              void* d_out, int out_size, void* d_ws, size_t ws_size,
                              hipStream_t stream) {
    const float* pred_mz = (const float*)d_in[0];
    const float* mz      = (const float*)d_in[1];
    const float* inten   = (const float*)d_in[2];

    float* inter = (float*)d_ws;           // 64 floats
    int*   pcnt  = (int*)d_ws + B_DIM;     // 64 ints
    int*   tcnt  = (int*)d_ws + 2 * B_DIM; // 64 ints

    init_ws_kernel<<<1, 256, 0, stream>>>(inter, pcnt, tcnt);
    count_kernel<<<(B_DIM * S_DIM) / 256, 256, 0, stream>>>(pred_mz, mz, pcnt, tcnt);
    soft_match_kernel<<<B_DIM * BLOCKS_PER_BATCH, 256, 0, stream>>>(
        pred_mz, mz, inten, inter);
    finalize_kernel<<<1, B_DIM, 0, stream>>>(inter, pcnt, tcnt, (float*)d_out);
}